// QuadricGrid_52295521796844
// MI455X (gfx1250) — compile-verified
//
#include <hip/hip_runtime.h>
#include <stdint.h>

#define LAYER 128

// ---- gfx1250 async-to-LDS path (guarded; falls back to plain loads) ------
#if defined(__AMDGCN__) && __has_builtin(__builtin_amdgcn_global_load_async_to_lds_b32)
#define HAVE_ASYNC_LDS 1
#else
#define HAVE_ASYNC_LDS 0
#endif

#if defined(__AMDGCN__) && __has_builtin(__builtin_amdgcn_s_wait_asynccnt)
#define WAIT_ASYNC0() __builtin_amdgcn_s_wait_asynccnt(0)
#else
#define WAIT_ASYNC0() asm volatile("s_wait_asynccnt 0" ::: "memory")
#endif

// Builtin signature (from round-1 diagnostic): param0 is AS1 (__device__) int*,
// so: (as1 int* gaddr, as3 int* ldsaddr, imm offset, imm cpol)
typedef __attribute__((address_space(1))) int* as1_int_ptr;
typedef __attribute__((address_space(3))) int* as3_int_ptr;

__global__ __launch_bounds__(256) void quadric_sdf_normal_kernel(
    const float* __restrict__ renderPts,   // n*3
    const int*   __restrict__ renderIdx,   // n
    const float* __restrict__ sdfPts,      // n*3
    const int*   __restrict__ sdfIdx,      // n
    const float* __restrict__ xL,          // 128
    const float* __restrict__ yL,          // 128
    const float* __restrict__ zL,          // 128
    const float* __restrict__ off,         // 4
    float* __restrict__ outSdf,            // n
    float* __restrict__ outNrm,            // n*3
    int n)
{
    __shared__ float s_x[LAYER];
    __shared__ float s_y[LAYER];
    __shared__ float s_z[LAYER];

    const int t = threadIdx.x;

#if HAVE_ASYNC_LDS
    if (t < LAYER) {
        // global side: integer round-trip (generic & AS1 global addresses are
        // bit-identical); LDS side: C-style addrspace cast to AS3.
        __builtin_amdgcn_global_load_async_to_lds_b32(
            (as1_int_ptr)(uintptr_t)(xL + t), (as3_int_ptr)(&s_x[t]), 0, 0);
        __builtin_amdgcn_global_load_async_to_lds_b32(
            (as1_int_ptr)(uintptr_t)(yL + t), (as3_int_ptr)(&s_y[t]), 0, 0);
        __builtin_amdgcn_global_load_async_to_lds_b32(
            (as1_int_ptr)(uintptr_t)(zL + t), (as3_int_ptr)(&s_z[t]), 0, 0);
    }
    WAIT_ASYNC0();   // drain this wave's ASYNCcnt before the barrier
#else
    if (t < LAYER) {
        s_x[t] = xL[t];
        s_y[t] = yL[t];
        s_z[t] = zL[t];
    }
#endif
    __syncthreads();

    const int i = (int)(blockIdx.x * blockDim.x + t);
    if (i >= n) return;

    // offset is uniform -> scalar loads
    const float l0 = off[0], l1 = off[1], l2 = off[2], cc = off[3];

    // ---------------- SDF stream ----------------
    {
        const int idx = __builtin_nontemporal_load(sdfIdx + i);
        const int ix = (idx >> 14) & (LAYER - 1);
        const int iy = (idx >> 7)  & (LAYER - 1);
        const int iz =  idx        & (LAYER - 1);
        const float qx = s_x[ix], qy = s_y[iy], qz = s_z[iz];

        const float px = __builtin_nontemporal_load(sdfPts + 3 * i + 0);
        const float py = __builtin_nontemporal_load(sdfPts + 3 * i + 1);
        const float pz = __builtin_nontemporal_load(sdfPts + 3 * i + 2);

        // (qx*px + l0)*px + (qy*py + l1)*py + (qz*pz + l2)*pz + c
        const float s =
            fmaf(fmaf(qx, px, l0), px,
            fmaf(fmaf(qy, py, l1), py,
            fmaf(fmaf(qz, pz, l2), pz, cc)));
        __builtin_nontemporal_store(s, outSdf + i);
    }

    // ---------------- Normal stream ----------------
    {
        const int idx = __builtin_nontemporal_load(renderIdx + i);
        const int ix = (idx >> 14) & (LAYER - 1);
        const int iy = (idx >> 7)  & (LAYER - 1);
        const int iz =  idx        & (LAYER - 1);
        const float qx = s_x[ix], qy = s_y[iy], qz = s_z[iz];

        const float px = __builtin_nontemporal_load(renderPts + 3 * i + 0);
        const float py = __builtin_nontemporal_load(renderPts + 3 * i + 1);
        const float pz = __builtin_nontemporal_load(renderPts + 3 * i + 2);

        const float gx = fmaf(2.0f * qx, px, l0);
        const float gy = fmaf(2.0f * qy, py, l1);
        const float gz = fmaf(2.0f * qz, pz, l2);
        const float inv =
            rsqrtf(fmaf(gx, gx, fmaf(gy, gy, fmaf(gz, gz, 1e-12f))));

        __builtin_nontemporal_store(gx * inv, outNrm + 3 * i + 0);
        __builtin_nontemporal_store(gy * inv, outNrm + 3 * i + 1);
        __builtin_nontemporal_store(gz * inv, outNrm + 3 * i + 2);
    }
}

extern "C" void kernel_launch(void* const* d_in, const int* in_sizes, int n_in,
                              void* d_out, int out_size, void* d_ws, size_t ws_size,
                              hipStream_t stream) {
    (void)n_in; (void)d_ws; (void)ws_size; (void)out_size;
    const float* renderPts = (const float*)d_in[0];
    const int*   renderIdx = (const int*)  d_in[1];
    const float* sdfPts    = (const float*)d_in[2];
    const int*   sdfIdx    = (const int*)  d_in[3];
    const float* xL        = (const float*)d_in[4];
    const float* yL        = (const float*)d_in[5];
    const float* zL        = (const float*)d_in[6];
    const float* off       = (const float*)d_in[7];

    const int n = in_sizes[1];          // element count of renderIndexList
    float* outSdf = (float*)d_out;      // [0, n)
    float* outNrm = outSdf + n;         // [n, 4n) as (n,3) row-major

    const int block = 256;              // 8 wave32 waves
    const int grid  = (n + block - 1) / block;
    quadric_sdf_normal_kernel<<<grid, block, 0, stream>>>(
        renderPts, renderIdx, sdfPts, sdfIdx, xL, yL, zL, off,
        outSdf, outNrm, n);
}